// BN_38027640438785
// MI455X (gfx1250) — compile-verified
//
#include <hip/hip_runtime.h>

// Column-L2-normalize:  out = x * rsqrt(sum_rows(x*x))   x: [N=524288, D=256] f32
//
// HBM-bound: 1.5 GB total traffic -> ~66us floor at 23.3 TB/s. Saturating 23.3 TB/s
// at ~600ns latency needs ~14MB in flight, so both passes are built around MLP:
//   Pass 1: 16384 waves, each issuing an 8-load clause per iteration feeding FOUR
//           independent V_WMMA_F32_16X16X4_F32 accumulator chains
//           (colsum(X^2) == ones(16x4)^T * (X^2)(4x16)).
//   Pass 2: float4 streaming scale, unrolled x2 (two B128 loads in flight per lane),
//           rsqrt factors staged in LDS, global_prefetch_b8 ahead of the stream.

typedef __attribute__((ext_vector_type(2))) float v2f;
typedef __attribute__((ext_vector_type(8))) float v8f;

#define DCOLS 256
#define RED_BLOCKS 2048   // 2048 blocks * 8 waves = 16384 waves = 16 col-tiles * 1024 row-groups
#define SCALE_BLOCKS 4096

__global__ void BN_zero_sums(float* __restrict__ sums) {
    sums[threadIdx.x] = 0.0f;   // <<<1,256>>>
}

__global__ __launch_bounds__(256) void BN_colsumsq_wmma(const float* __restrict__ x,
                                                        float* __restrict__ sums,
                                                        int nrows) {
    const int lane     = threadIdx.x & 31;
    const int wave     = (blockIdx.x * blockDim.x + threadIdx.x) >> 5;
    const int numWaves = (gridDim.x * blockDim.x) >> 5;

    const int colTile      = wave & 15;        // 16 tiles x 16 columns = 256 columns
    const int rowGroup     = wave >> 4;
    const int numRowGroups = numWaves >> 4;    // 1024
    const int col0         = colTile << 4;

    // B-matrix (4x16 f32, KxN, 2 VGPRs, wave32), symmetric to documented A 16x4 layout:
    //   VGPR0: lanes 0-15 -> (K=0, N=lane);  lanes 16-31 -> (K=2, N=lane-16)
    //   VGPR1: lanes 0-15 -> (K=1, N=lane);  lanes 16-31 -> (K=3, N=lane-16)
    const int n     = lane & 15;
    const int kBase = (lane >> 4) << 1;        // 0 or 2

    v8f a0 = {}, a1 = {}, a2 = {}, a3 = {};    // 4 independent XDL accumulator chains
    v2f ones; ones.x = 1.0f; ones.y = 1.0f;    // A = 16x4 ones -> D[m,n] = sum_k B[k,n] + C[m,n]

    const float*  base    = x + col0 + n;
    const int     hopRows = numRowGroups * 16; // rows consumed grid-wide per iteration
    const size_t  loopHop = (size_t)hopRows * DCOLS;

    for (int r = rowGroup * 16; r + 15 < nrows; r += hopRows) {
        const float* p = base + (size_t)(r + kBase) * DCOLS;
        __builtin_prefetch(p + loopHop, 0, 0);          // global_prefetch_b8
        // 8 loads, one base VGPR pair, immediate offsets -> single s_clause, 1KB in flight
        float b0 = p[0 * DCOLS], b1 = p[1 * DCOLS];     // rows r+kBase,    +1
        float c0 = p[4 * DCOLS], c1 = p[5 * DCOLS];     // rows r+4+kBase,  +1
        float d0 = p[8 * DCOLS], d1 = p[9 * DCOLS];     // rows r+8+kBase,  +1
        float e0 = p[12 * DCOLS], e1 = p[13 * DCOLS];   // rows r+12+kBase, +1
        v2f b; b.x = b0 * b0; b.y = b1 * b1;            // VALU squaring co-executes with XDL
        v2f c; c.x = c0 * c0; c.y = c1 * c1;
        v2f d; d.x = d0 * d0; d.y = d1 * d1;
        v2f e; e.x = e0 * e0; e.y = e1 * e1;
        a0 = __builtin_amdgcn_wmma_f32_16x16x4_f32(false, ones, false, b, (short)0, a0, false, false);
        a1 = __builtin_amdgcn_wmma_f32_16x16x4_f32(false, ones, false, c, (short)0, a1, false, false);
        a2 = __builtin_amdgcn_wmma_f32_16x16x4_f32(false, ones, false, d, (short)0, a2, false, false);
        a3 = __builtin_amdgcn_wmma_f32_16x16x4_f32(false, ones, false, e, (short)0, a3, false, false);
    }

    // C/D layout: VGPR0, lanes 0-15 hold row M=0 (N = lane). Add exactly one row.
    const float s = (a0[0] + a1[0]) + (a2[0] + a3[0]);
    if (lane < 16) atomicAdd(&sums[col0 + n], s);
}

__global__ __launch_bounds__(256) void BN_scale(const float4* __restrict__ x4,
                                                float4* __restrict__ out4,
                                                const float* __restrict__ sums,
                                                size_t n4) {
    __shared__ float inv[DCOLS];
    inv[threadIdx.x] = __frsqrt_rn(sums[threadIdx.x]);
    __syncthreads();

    const size_t stride  = (size_t)gridDim.x * blockDim.x;
    const size_t stride2 = stride * 2;
    size_t i = (size_t)blockIdx.x * blockDim.x + threadIdx.x;

    // Unroll x2: both B128 loads issued before either result is consumed.
    for (; i + stride < n4; i += stride2) {
        __builtin_prefetch(&x4[i + stride2], 0, 0);
        const size_t j  = i + stride;
        const int    ca = (int)(i & 63) << 2;  // first column of this float4: (4*i) % 256
        const int    cb = (int)(j & 63) << 2;
        float4 va = x4[i];
        float4 vb = x4[j];
        va.x *= inv[ca + 0]; va.y *= inv[ca + 1]; va.z *= inv[ca + 2]; va.w *= inv[ca + 3];
        vb.x *= inv[cb + 0]; vb.y *= inv[cb + 1]; vb.z *= inv[cb + 2]; vb.w *= inv[cb + 3];
        out4[i] = va;
        out4[j] = vb;
    }
    if (i < n4) {                              // generic tail (not hit for this shape)
        const int c = (int)(i & 63) << 2;
        float4 v = x4[i];
        v.x *= inv[c + 0]; v.y *= inv[c + 1]; v.z *= inv[c + 2]; v.w *= inv[c + 3];
        out4[i] = v;
    }
}

extern "C" void kernel_launch(void* const* d_in, const int* in_sizes, int n_in,
                              void* d_out, int out_size, void* d_ws, size_t ws_size,
                              hipStream_t stream) {
    const float* x    = (const float*)d_in[0];
    float*       out  = (float*)d_out;
    float*       sums = (float*)d_ws;          // 256 floats of scratch

    const long long total = in_sizes[0];       // 524288 * 256
    const int nrows = (int)(total / DCOLS);
    const size_t n4 = (size_t)total / 4;

    BN_zero_sums<<<1, DCOLS, 0, stream>>>(sums);
    BN_colsumsq_wmma<<<RED_BLOCKS, 256, 0, stream>>>(x, sums, nrows);
    BN_scale<<<SCALE_BLOCKS, 256, 0, stream>>>((const float4*)x, (float4*)out, sums, n4);
}